// XCA_43989055045944
// MI455X (gfx1250) — compile-verified
//
#include <hip/hip_runtime.h>
#include <hip/hip_bf16.h>
#include <math.h>

// ---------------------------------------------------------------------------
// XCA (cross-covariance attention), fp32, on gfx1250 using V_WMMA_F32_16X16X4_F32
// B=8, N=4096, D=768, H=16, C=48
// LDS tiles are K-pair interleaved so each WMMA fragment is one ds_load_b64.
// ---------------------------------------------------------------------------

typedef __attribute__((ext_vector_type(2))) float v2f;
typedef __attribute__((ext_vector_type(8))) float v8f;

__device__ __forceinline__ v8f wmma4(v2f a, v2f b, v8f c) {
  // D = A(16x4 f32) * B(4x16 f32) + C(16x16 f32)
  return __builtin_amdgcn_wmma_f32_16x16x4_f32(
      /*neg_a=*/false, a, /*neg_b=*/false, b,
      /*c_mod=*/(short)0, c, /*reuse_a=*/false, /*reuse_b=*/false);
}

// ---------------------------------------------------------------------------
// Generic fp32 GEMM: C[M,N] = A[M,K](lda) * B[K,N](ldb) (+bias[N])
// Block tile 128x64, 8 waves, each wave 32x32 (2x2 WMMA tiles), BK=16.
// Double-buffered LDS; pair-interleaved layout:
//   As2[buf][p][m*2+j] = A[m][kb+2p+j]   (p = 0..7, row stride 288: delta 32 banks)
//   Bs2[buf][p][n*2+j] = B[kb+2p+j][n]   (row stride 160: delta 32 banks)
// ---------------------------------------------------------------------------
#define BM 128
#define BN 64
#define BK 16
#define ASP 288   // >= 256; 288 % 64 == 32 -> hi half-wave on disjoint banks
#define BSP 160   // >= 128; 160 % 64 == 32

__global__ __launch_bounds__(256) void gemm_f32_wmma(
    const float* __restrict__ A, int lda,
    const float* __restrict__ B, int ldb,
    const float* __restrict__ bias,
    float* __restrict__ C, int ldc,
    int M, int N, int K)
{
  __shared__ float As2[2][BK / 2][ASP];
  __shared__ float Bs2[2][BK / 2][BSP];

  const int tid  = threadIdx.x;
  const int wid  = tid >> 5;
  const int lane = tid & 31;
  const int ll   = lane & 15;
  const int hi   = lane >> 4;

  const int bm = blockIdx.y * BM;
  const int bn = blockIdx.x * BN;
  const int wm = (wid >> 1) * 32;   // 0/32/64/96
  const int wn = (wid & 1) * 32;    // 0/32

  v8f acc[2][2];
#pragma unroll
  for (int i = 0; i < 2; ++i)
#pragma unroll
    for (int j = 0; j < 2; ++j)
#pragma unroll
      for (int e = 0; e < 8; ++e) acc[i][j][e] = 0.0f;

  // staging assignments
  const int akp = tid & 3;          // k quarter (4 consecutive k) for A
  const int am  = tid >> 2;         // 0..63 (two m-halves)
  const int bp  = tid >> 5;         // 0..7 k-pair for B
  const int bn2 = (tid & 31) * 2;   // n pair base

  // ---- stage slab kb into LDS buffer `buf` ----
  auto stage = [&](int kb, int buf) {
#pragma unroll
    for (int half = 0; half < 2; ++half) {
      const int m = am + half * 64;
      const float4 a4 = *reinterpret_cast<const float4*>(
          &A[(size_t)(bm + m) * lda + kb + akp * 4]);
      float2 p0; p0.x = a4.x; p0.y = a4.y;
      float2 p1; p1.x = a4.z; p1.y = a4.w;
      *reinterpret_cast<float2*>(&As2[buf][akp * 2 + 0][m * 2]) = p0;
      *reinterpret_cast<float2*>(&As2[buf][akp * 2 + 1][m * 2]) = p1;
    }
    const float2 g0 = *reinterpret_cast<const float2*>(
        &B[(size_t)(kb + 2 * bp + 0) * ldb + bn + bn2]);
    const float2 g1 = *reinterpret_cast<const float2*>(
        &B[(size_t)(kb + 2 * bp + 1) * ldb + bn + bn2]);
    float2 q0; q0.x = g0.x; q0.y = g1.x;   // (B[k][n], B[k+1][n])
    float2 q1; q1.x = g0.y; q1.y = g1.y;   // (B[k][n+1], B[k+1][n+1])
    *reinterpret_cast<float2*>(&Bs2[buf][bp][bn2 * 2 + 0]) = q0;
    *reinterpret_cast<float2*>(&Bs2[buf][bp][bn2 * 2 + 2]) = q1;
  };

  stage(0, 0);
  __syncthreads();

  int buf = 0;
  for (int kb = 0; kb < K; kb += BK) {
    if (kb + BK < K) {
      stage(kb + BK, buf ^ 1);      // overlaps with WMMA on current buffer
      __builtin_prefetch(&A[(size_t)(bm + am) * lda + kb + 2 * BK], 0, 0);
      __builtin_prefetch(&B[(size_t)(kb + 2 * BK + bp) * ldb + bn + bn2], 0, 0);
    }
#pragma unroll
    for (int ks = 0; ks < BK; ks += 4) {
      const int p = (ks >> 1) + hi;   // k-pair index for this half-wave
      v2f af[2], bf[2];
#pragma unroll
      for (int t = 0; t < 2; ++t) {
        af[t] = *reinterpret_cast<const v2f*>(&As2[buf][p][(wm + t * 16 + ll) * 2]);
        bf[t] = *reinterpret_cast<const v2f*>(&Bs2[buf][p][(wn + t * 16 + ll) * 2]);
      }
#pragma unroll
      for (int ti = 0; ti < 2; ++ti)
#pragma unroll
        for (int tj = 0; tj < 2; ++tj)
          acc[ti][tj] = wmma4(af[ti], bf[tj], acc[ti][tj]);
    }
    __syncthreads();
    buf ^= 1;
  }

  // epilogue: D vgpr r holds rows r (lanes 0-15) and r+8 (lanes 16-31)
#pragma unroll
  for (int ti = 0; ti < 2; ++ti) {
#pragma unroll
    for (int tj = 0; tj < 2; ++tj) {
      const int col = bn + wn + tj * 16 + ll;
      const float bv = bias ? bias[col] : 0.0f;
#pragma unroll
      for (int r = 0; r < 8; ++r) {
        const int row = bm + wm + ti * 16 + r + 8 * hi;
        C[(size_t)row * ldc + col] = acc[ti][tj][r] + bv;
      }
    }
  }
}

// ---------------------------------------------------------------------------
// Fused XCA core: one block per (b,h).
//   pass1: S[48,48] = q.kT over N (WMMA), fused L2-norm sums of q,k rows
//   softmax(S * invq * invk * temperature)
//   pass2: out = P @ V, written in place over the q region of qkv
// qkv layout: [B, N, 3*D], q at col h*48, k at 768+h*48, v at 1536+h*48
// LDS chunks pair-interleaved over n: Qp[p][c*2+j] = q[c][n0+2p+j]
//   (pair-row stride 96 floats: 96 % 64 == 32 -> hi half-wave bank-disjoint)
// ---------------------------------------------------------------------------
#define XN    4096
#define XTD   2304
#define XC    48
#define CHUNK 64
#define QP    (2 * XC)   // 96

__global__ __launch_bounds__(288) void xca_attn(
    float* __restrict__ qkv, const float* __restrict__ temperature)
{
  __shared__ float Qp[CHUNK / 2][QP];   // reused as Vp in pass 2
  __shared__ float Kp[CHUNK / 2][QP];
  __shared__ float Ps[XC][XC + 1];
  __shared__ float invq[XC];
  __shared__ float invk[XC];

  const int b = blockIdx.x >> 4;
  const int h = blockIdx.x & 15;
  const size_t base = (size_t)b * XN * XTD;
  const int qoff = h * XC;
  const int koff = 768 + h * XC;
  const int voff = 1536 + h * XC;

  const int tid  = threadIdx.x;
  const int wid  = tid >> 5;           // 0..8
  const int lane = tid & 31;
  const int ll   = lane & 15;
  const int hi   = lane >> 4;
  const int ci   = wid / 3;            // S row tile (0..2)
  const int di   = wid % 3;            // S col tile (0..2)

  v8f acc;
#pragma unroll
  for (int e = 0; e < 8; ++e) acc[e] = 0.0f;

  float sumsq = 0.0f;                  // meaningful for tid < 96

  // ---- pass 1: accumulate S and norm sums over N ----
  for (int n0 = 0; n0 < XN; n0 += CHUNK) {
    for (int idx = tid; idx < CHUNK * XC; idx += 288) {
      const int r = idx / XC;
      const int c = idx - r * XC;
      const size_t row = base + (size_t)(n0 + r) * XTD;
      Qp[r >> 1][c * 2 + (r & 1)] = qkv[row + qoff + c];
      Kp[r >> 1][c * 2 + (r & 1)] = qkv[row + koff + c];
    }
    __syncthreads();

    if (tid < XC) {
#pragma unroll 8
      for (int p = 0; p < CHUNK / 2; ++p) {
        const float v0 = Qp[p][tid * 2], v1 = Qp[p][tid * 2 + 1];
        sumsq += v0 * v0 + v1 * v1;
      }
    } else if (tid < 2 * XC) {
      const int c = tid - XC;
#pragma unroll 8
      for (int p = 0; p < CHUNK / 2; ++p) {
        const float v0 = Kp[p][c * 2], v1 = Kp[p][c * 2 + 1];
        sumsq += v0 * v0 + v1 * v1;
      }
    }

#pragma unroll
    for (int ks = 0; ks < CHUNK; ks += 4) {
      const int p = (ks >> 1) + hi;
      const v2f a  = *reinterpret_cast<const v2f*>(&Qp[p][(ci * 16 + ll) * 2]);
      const v2f bb = *reinterpret_cast<const v2f*>(&Kp[p][(di * 16 + ll) * 2]);
      acc = wmma4(a, bb, acc);
    }
    __syncthreads();
  }

  if (tid < XC)            invq[tid]      = 1.0f / fmaxf(sqrtf(sumsq), 1e-12f);
  else if (tid < 2 * XC)   invk[tid - XC] = 1.0f / fmaxf(sqrtf(sumsq), 1e-12f);

  // spill S tiles to LDS
#pragma unroll
  for (int r = 0; r < 8; ++r)
    Ps[ci * 16 + r + 8 * hi][di * 16 + ll] = acc[r];
  __syncthreads();

  // ---- softmax over last axis (rows of length 48), scaled ----
  if (tid < XC) {
    const float tmp = temperature[h];
    const float iq  = invq[tid];
    float mx = -3.4e38f;
    for (int d = 0; d < XC; ++d) {
      const float v = Ps[tid][d] * iq * invk[d] * tmp;
      Ps[tid][d] = v;
      mx = fmaxf(mx, v);
    }
    float s = 0.0f;
    for (int d = 0; d < XC; ++d) {
      const float e = __expf(Ps[tid][d] - mx);
      Ps[tid][d] = e;
      s += e;
    }
    const float inv = 1.0f / s;
    for (int d = 0; d < XC; ++d) Ps[tid][d] *= inv;
  }
  __syncthreads();

  // ---- pass 2: out[c,n] = sum_d P[c,d] * v[d,n], in place over q region ----
  for (int n0 = 0; n0 < XN; n0 += CHUNK) {
    for (int idx = tid; idx < CHUNK * XC; idx += 288) {
      const int r = idx / XC;
      const int c = idx - r * XC;
      Qp[r >> 1][c * 2 + (r & 1)] =
          qkv[base + (size_t)(n0 + r) * XTD + voff + c];   // V chunk
    }
    __syncthreads();

    const int c = tid % XC;
    for (int nn = tid / XC; nn < CHUNK; nn += 6) {
      const int p = nn >> 1, j = nn & 1;
      float o = 0.0f;
#pragma unroll 8
      for (int d = 0; d < XC; ++d) o += Ps[c][d] * Qp[p][d * 2 + j];
      qkv[base + (size_t)(n0 + nn) * XTD + qoff + c] = o;
    }
    __syncthreads();
  }
}

// ---------------------------------------------------------------------------
// Host launcher
// ---------------------------------------------------------------------------
extern "C" void kernel_launch(void* const* d_in, const int* in_sizes, int n_in,
                              void* d_out, int out_size, void* d_ws, size_t ws_size,
                              hipStream_t stream) {
  const float* x           = (const float*)d_in[0];   // [8,4096,768]
  const float* w_qkv       = (const float*)d_in[1];   // [768,2304]
  const float* temperature = (const float*)d_in[2];   // [16]
  const float* w_proj      = (const float*)d_in[3];   // [768,768]
  const float* b_proj      = (const float*)d_in[4];   // [768]
  float* out = (float*)d_out;                         // [8,4096,768]
  float* qkv = (float*)d_ws;                          // [8,4096,2304] fp32

  const int M  = 8 * 4096;   // 32768 rows
  const int D  = 768;
  const int N3 = 3 * D;      // 2304

  // 1) QKV GEMM: qkv = x @ w_qkv
  {
    dim3 grid(N3 / BN, M / BM);   // 36 x 256
    gemm_f32_wmma<<<grid, 256, 0, stream>>>(x, D, w_qkv, N3, nullptr,
                                            qkv, N3, M, N3, D);
  }

  // 2) Fused normalize + XCA attention + P@V (in place into q region)
  xca_attn<<<8 * 16, 288, 0, stream>>>(qkv, temperature);

  // 3) Projection GEMM: out = attn_out @ w_proj + b_proj   (lda = 2304!)
  {
    dim3 grid(D / BN, M / BM);    // 12 x 256
    gemm_f32_wmma<<<grid, 256, 0, stream>>>(qkv, N3, w_proj, D, b_proj,
                                            out, D, M, D, D);
  }
}